// SimpleGCN2_23965917512419
// MI455X (gfx1250) — compile-verified
//
#include <hip/hip_runtime.h>

typedef __attribute__((ext_vector_type(2))) float v2f;
typedef __attribute__((ext_vector_type(8))) float v8f;

#define F_IN 128
#define HID  16
#define NCLS 7

// ---------------- degree / norm ----------------

__global__ void k_init_deg(float* __restrict__ deg, int N) {
    int i = blockIdx.x * blockDim.x + threadIdx.x;
    if (i < N) deg[i] = 1.0f;   // self-loop contribution
}

__global__ void k_deg_accum(const long long* __restrict__ dst,
                            float* __restrict__ deg, long long E) {
    long long e = (long long)blockIdx.x * blockDim.x + threadIdx.x;
    if (e < E) atomicAdd(&deg[(int)dst[e]], 1.0f);
}

// ---------------- layer 1 GEMM: h1 = x @ W1 (WMMA f32 16x16x4) ----------------
// One wave computes a 16(M) x 16(N=HID) tile, K=128 via 32 x4-WMMAs.
// W1 (8 KB) is staged once per block into LDS; B fragments come from ds loads.

__global__ void k_gemm1(const float* __restrict__ x, const float* __restrict__ W1,
                        float* __restrict__ h1, int N) {
    __shared__ float sW1[F_IN * HID];

    // cooperative stage: 2048 floats as 512 float4s
    for (int idx = threadIdx.x; idx < (F_IN * HID) / 4; idx += blockDim.x)
        ((float4*)sW1)[idx] = ((const float4*)W1)[idx];
    __syncthreads();

    int wave = threadIdx.x >> 5;
    int lane = threadIdx.x & 31;
    int tile = blockIdx.x * (blockDim.x >> 5) + wave;
    int rowBase = tile * 16;
    if (rowBase >= N) return;                  // wave-uniform (EXEC all-1s for WMMA)

    int half = lane >> 4;                      // 0: K pair {0,1}, 1: K pair {2,3}
    int l    = lane & 15;                      // A: row M=l ; B/C: col N=l
    int arow = (rowBase + l < N) ? (rowBase + l) : (N - 1);
    const float* xrow = x + (long long)arow * F_IN;

    v8f c = {};
#pragma unroll 4
    for (int kk = 0; kk < F_IN / 4; ++kk) {
        int koff = kk * 4 + half * 2;
        v2f a = *(const v2f*)(xrow + koff);    // single b64 load
        v2f b;
        b.x = sW1[koff * HID + l];
        b.y = sW1[(koff + 1) * HID + l];
        c = __builtin_amdgcn_wmma_f32_16x16x4_f32(false, a, false, b,
                                                  (short)0, c, false, false);
    }

    if (rowBase + 16 <= N) {                   // fast path: unguarded stores
#pragma unroll
        for (int r = 0; r < 8; ++r)
            h1[(long long)(rowBase + r + half * 8) * HID + l] = c[r];
    } else {                                   // tail tile
#pragma unroll
        for (int r = 0; r < 8; ++r) {
            int m = rowBase + r + half * 8;
            if (m < N) h1[(long long)m * HID + l] = c[r];
        }
    }
}

// ---------------- dinv + out1 init (bias + self-loop term) ----------------

__global__ void k_dinv_out1(float* __restrict__ degdinv,
                            const float* __restrict__ h1,
                            const float* __restrict__ b1,
                            float* __restrict__ out1, int N) {
    int i = blockIdx.x * blockDim.x + threadIdx.x;
    if (i >= N) return;
    float dv = rsqrtf(degdinv[i]);         // deg >= 1 always (self loop)
    degdinv[i] = dv;                       // buffer now holds dinv
    float s = dv * dv;
    const float4* hh = (const float4*)(h1 + (long long)i * HID);
    float4*       oo = (float4*)(out1 + (long long)i * HID);
#pragma unroll
    for (int q = 0; q < 4; ++q) {
        float4 h = hh[q];
        float4 r;
        r.x = b1[q * 4 + 0] + h.x * s;
        r.y = b1[q * 4 + 1] + h.y * s;
        r.z = b1[q * 4 + 2] + h.z * s;
        r.w = b1[q * 4 + 3] + h.w * s;
        oo[q] = r;
    }
}

// ---------------- layer 1 edge scatter: 4 features per thread ----------------

__global__ void k_scatter1(const long long* __restrict__ src,
                           const long long* __restrict__ dst,
                           const float* __restrict__ dinv,
                           const float* __restrict__ h1,
                           float* __restrict__ out1, long long E) {
    long long t = (long long)blockIdx.x * blockDim.x + threadIdx.x;
    long long e = t >> 2;
    int q = (int)(t & 3);                  // feature quad 0..3
    if (e >= E) return;
    int s = (int)src[e], d = (int)dst[e];
    float w = dinv[s] * dinv[d];
    float4 hv = *(const float4*)(h1 + (long long)s * HID + q * 4);
    float* o = out1 + (long long)d * HID + q * 4;
    atomicAdd(o + 0, hv.x * w);
    atomicAdd(o + 1, hv.y * w);
    atomicAdd(o + 2, hv.z * w);
    atomicAdd(o + 3, hv.w * w);
}

// ---------------- layer 2 GEMM: h2 = relu(out1) @ W2 (WMMA f32 16x16x4) ----------------
// W2 [16x7] zero-padded to 16 B-columns; ReLU fused into A load;
// bias + self-loop term fused into the out2 init store.

__global__ void k_gemm2(const float* __restrict__ out1, const float* __restrict__ W2,
                        const float* __restrict__ b2, const float* __restrict__ dinv,
                        float* __restrict__ h2, float* __restrict__ out2, int N) {
    int wave = threadIdx.x >> 5;
    int lane = threadIdx.x & 31;
    int tile = blockIdx.x * (blockDim.x >> 5) + wave;
    int rowBase = tile * 16;
    if (rowBase >= N) return;

    int half = lane >> 4;
    int l    = lane & 15;
    int arow = (rowBase + l < N) ? (rowBase + l) : (N - 1);
    const float* a0 = out1 + (long long)arow * HID;

    v8f c = {};
#pragma unroll
    for (int kk = 0; kk < HID / 4; ++kk) {
        int koff = kk * 4 + half * 2;
        v2f av = *(const v2f*)(a0 + koff);
        v2f a, b;
        a.x = fmaxf(av.x, 0.0f);
        a.y = fmaxf(av.y, 0.0f);
        b.x = (l < NCLS) ? W2[koff * NCLS + l] : 0.0f;
        b.y = (l < NCLS) ? W2[(koff + 1) * NCLS + l] : 0.0f;
        c = __builtin_amdgcn_wmma_f32_16x16x4_f32(false, a, false, b,
                                                  (short)0, c, false, false);
    }

    if (l < NCLS) {
        float bias = b2[l];
        if (rowBase + 16 <= N) {
#pragma unroll
            for (int r = 0; r < 8; ++r) {
                int node = rowBase + r + half * 8;
                float v = c[r];
                float dv = dinv[node];
                h2[(long long)node * NCLS + l]   = v;
                out2[(long long)node * NCLS + l] = bias + v * dv * dv;
            }
        } else {
#pragma unroll
            for (int r = 0; r < 8; ++r) {
                int node = rowBase + r + half * 8;
                if (node < N) {
                    float v = c[r];
                    float dv = dinv[node];
                    h2[(long long)node * NCLS + l]   = v;
                    out2[(long long)node * NCLS + l] = bias + v * dv * dv;
                }
            }
        }
    }
}

// ---------------- layer 2 edge scatter: one thread per edge ----------------

__global__ void k_scatter2(const long long* __restrict__ src,
                           const long long* __restrict__ dst,
                           const float* __restrict__ dinv,
                           const float* __restrict__ h2,
                           float* __restrict__ out2, long long E) {
    long long e = (long long)blockIdx.x * blockDim.x + threadIdx.x;
    if (e >= E) return;
    int s = (int)src[e], d = (int)dst[e];
    float w = dinv[s] * dinv[d];
    const float* hs = h2 + (long long)s * NCLS;
    float*       o  = out2 + (long long)d * NCLS;
#pragma unroll
    for (int cc = 0; cc < NCLS; ++cc)
        atomicAdd(o + cc, hs[cc] * w);
}

// ---------------- row-wise log_softmax over 7 classes ----------------

__global__ void k_logsm(const float* __restrict__ out2, float* __restrict__ y, int N) {
    int i = blockIdx.x * blockDim.x + threadIdx.x;
    if (i >= N) return;
    float v[NCLS];
    float m = -3.402823466e+38f;
#pragma unroll
    for (int c = 0; c < NCLS; ++c) {
        v[c] = out2[(long long)i * NCLS + c];
        m = fmaxf(m, v[c]);
    }
    float sum = 0.0f;
#pragma unroll
    for (int c = 0; c < NCLS; ++c) sum += expf(v[c] - m);
    float lse = logf(sum);
#pragma unroll
    for (int c = 0; c < NCLS; ++c)
        y[(long long)i * NCLS + c] = v[c] - m - lse;
}

// ---------------- launcher ----------------

extern "C" void kernel_launch(void* const* d_in, const int* in_sizes, int n_in,
                              void* d_out, int out_size, void* d_ws, size_t ws_size,
                              hipStream_t stream) {
    const float*     x   = (const float*)d_in[0];
    const long long* ei  = (const long long*)d_in[1];
    const float*     W1  = (const float*)d_in[2];
    const float*     b1  = (const float*)d_in[3];
    const float*     W2  = (const float*)d_in[4];
    const float*     b2  = (const float*)d_in[5];
    float*           y   = (float*)d_out;

    const int       N = in_sizes[0] / F_IN;
    const long long E = (long long)in_sizes[1] / 2;
    const long long* src = ei;
    const long long* dst = ei + E;

    // workspace layout (floats): dinv[N] | h1[16N] | out1[16N] | h2[7N] | out2[7N]
    float* ws    = (float*)d_ws;
    float* dinv  = ws;
    float* h1    = ws + (long long)N;
    float* out1  = ws + (long long)17 * N;
    float* h2    = ws + (long long)33 * N;
    float* out2  = ws + (long long)40 * N;

    const int TB = 256;
    int gN = (N + TB - 1) / TB;

    // 1) deg = 1 (self loops)
    k_init_deg<<<gN, TB, 0, stream>>>(dinv, N);
    // 2) deg += indegree
    {
        long long g = (E + TB - 1) / TB;
        k_deg_accum<<<(unsigned)g, TB, 0, stream>>>(dst, dinv, E);
    }
    // 3) h1 = x @ W1   (WMMA, W1 staged in LDS)
    {
        int tiles = (N + 15) / 16;
        int wavesPerBlock = TB / 32;
        int g = (tiles + wavesPerBlock - 1) / wavesPerBlock;
        k_gemm1<<<g, TB, 0, stream>>>(x, W1, h1, N);
    }
    // 4) dinv = rsqrt(deg); out1 = b1 + h1 * dinv^2  (self-loop term)
    k_dinv_out1<<<gN, TB, 0, stream>>>(dinv, h1, b1, out1, N);
    // 5) scatter layer 1 (4 features per thread)
    {
        long long total = E * 4;
        long long g = (total + TB - 1) / TB;
        k_scatter1<<<(unsigned)g, TB, 0, stream>>>(src, dst, dinv, h1, out1, E);
    }
    // 6) h2 = relu(out1) @ W2 ; out2 = b2 + h2 * dinv^2  (WMMA)
    {
        int tiles = (N + 15) / 16;
        int wavesPerBlock = TB / 32;
        int g = (tiles + wavesPerBlock - 1) / wavesPerBlock;
        k_gemm2<<<g, TB, 0, stream>>>(out1, W2, b2, dinv, h2, out2, N);
    }
    // 7) scatter layer 2 (one thread per edge)
    {
        long long g = (E + TB - 1) / TB;
        k_scatter2<<<(unsigned)g, TB, 0, stream>>>(src, dst, dinv, h2, out2, E);
    }
    // 8) log_softmax -> d_out
    k_logsm<<<gN, TB, 0, stream>>>(out2, y, N);
}